// GNNEncoder_50766513438800
// MI455X (gfx1250) — compile-verified
//
#include <hip/hip_runtime.h>
#include <hip/hip_bf16.h>
#include <math.h>

typedef __attribute__((ext_vector_type(2))) float v2f;
typedef __attribute__((ext_vector_type(8))) float v8f;

#define HID 96
#define HEADS 4
#define DH 24
#define NEG_SLOPE 0.2f
#define MAXK 128
#define LDSW 104   // padded LDS row stride: 2*104 mod 64 = 16 -> halves hit disjoint banks

// ---------------------------------------------------------------------------
// FP32 WMMA GEMM: C[M x 96] = A[M x K] * B[K x 96]
// W is staged once per workgroup into LDS (padded stride kills bank conflicts),
// then each wave computes a 16-row x 96-col strip with 6 f32 WMMA accumulators,
// reusing the A fragment across the 6 column tiles per k-step.
// Requires M % 16 == 0, K % 4 == 0 (true here: M=50000, K=128 or 96).
// ---------------------------------------------------------------------------
__global__ void gat_gemm_wmma(const float* __restrict__ A,
                              const float* __restrict__ B,
                              float* __restrict__ C,
                              int M, int K) {
    __shared__ float sW[MAXK * LDSW];

    // cooperative stage of W into LDS (all waves participate BEFORE any exit)
    for (int i = threadIdx.x; i < K * HID; i += blockDim.x) {
        const int r = i / HID, c = i - r * HID;
        sW[r * LDSW + c] = B[i];
    }
    __syncthreads();

    const int wavesPerBlock = blockDim.x >> 5;
    const int mt = blockIdx.x * wavesPerBlock + (threadIdx.x >> 5);   // 16-row tile
    if (mt * 16 >= M) return;                                         // whole-wave exit

    const int lane = threadIdx.x & 31;
    const int lr   = lane & 15;        // row (A) / col (B,C)
    const int half = lane >> 4;        // selects K pair / M half

    v8f acc[6] = {v8f{}, v8f{}, v8f{}, v8f{}, v8f{}, v8f{}};

    const int row = mt * 16 + lr;
    for (int k = 0; k < K; k += 4) {
        const int ka = k + half * 2;
        v2f a;
        a.x = A[row * K + ka];
        a.y = A[row * K + ka + 1];
#pragma unroll
        for (int j = 0; j < 6; ++j) {
            v2f b;
            b.x = sW[ka * LDSW + j * 16 + lr];
            b.y = sW[(ka + 1) * LDSW + j * 16 + lr];
            acc[j] = __builtin_amdgcn_wmma_f32_16x16x4_f32(
                false, a, false, b, (short)0, acc[j], false, false);
        }
    }

    // 16x16 f32 D layout: VGPR i holds M = i (lanes 0-15) / M = i+8 (lanes 16-31)
#pragma unroll
    for (int j = 0; j < 6; ++j) {
#pragma unroll
        for (int i = 0; i < 8; ++i) {
            const int r = mt * 16 + half * 8 + i;
            C[r * HID + j * 16 + lr] = acc[j][i];
        }
    }
}

// ---------------------------------------------------------------------------
// Per-node attention logits: al[n][h] = sum_d h[n][h*DH+d] * a[h][d]
// ---------------------------------------------------------------------------
__global__ void gat_logits(const float* __restrict__ h,
                           const float* __restrict__ a_src,
                           const float* __restrict__ a_dst,
                           float* __restrict__ alsrc,
                           float* __restrict__ aldst, int N) {
    int idx = blockIdx.x * blockDim.x + threadIdx.x;  // n*HEADS + head
    if (idx >= N * HEADS) return;
    const int n  = idx >> 2;
    const int hh = idx & 3;
    const float* hp = h + n * HID + hh * DH;
    float s = 0.f, d = 0.f;
#pragma unroll
    for (int i = 0; i < DH; ++i) {
        const float v = hp[i];
        s += v * a_src[hh * DH + i];
        d += v * a_dst[hh * DH + i];
    }
    alsrc[idx] = s;
    aldst[idx] = d;
}

__global__ void fill_f32(float* __restrict__ p, float v, int n) {
    int i = blockIdx.x * blockDim.x + threadIdx.x;
    if (i < n) p[i] = v;
}

__device__ __forceinline__ float leaky(float v) {
    return v > 0.f ? v : NEG_SLOPE * v;
}

// Monotonic-bits float atomic max (init must be -inf).
__device__ __forceinline__ void atomicMaxF(float* addr, float val) {
    if (val >= 0.f)
        atomicMax((int*)addr, __float_as_int(val));
    else
        atomicMin((unsigned int*)addr, __float_as_uint(val));
}

// Pass 1: segment max of leaky_relu(alsrc[src]+aldst[dst]) over dst
__global__ void gat_edge_max(const int* __restrict__ src,
                             const int* __restrict__ dst,
                             const float* __restrict__ alsrc,
                             const float* __restrict__ aldst,
                             float* __restrict__ mx, int E) {
    int e = blockIdx.x * blockDim.x + threadIdx.x;
    if (e >= E) return;
    const int s = src[e], d = dst[e];
#pragma unroll
    for (int hh = 0; hh < HEADS; ++hh) {
        const float v = leaky(alsrc[s * HEADS + hh] + aldst[d * HEADS + hh]);
        atomicMaxF(&mx[d * HEADS + hh], v);
    }
}

// Pass 2: denom[dst] += exp(e - mx[dst])
__global__ void gat_edge_sum(const int* __restrict__ src,
                             const int* __restrict__ dst,
                             const float* __restrict__ alsrc,
                             const float* __restrict__ aldst,
                             const float* __restrict__ mx,
                             float* __restrict__ den, int E) {
    int e = blockIdx.x * blockDim.x + threadIdx.x;
    if (e >= E) return;
    const int s = src[e], d = dst[e];
#pragma unroll
    for (int hh = 0; hh < HEADS; ++hh) {
        const float v = leaky(alsrc[s * HEADS + hh] + aldst[d * HEADS + hh]);
        atomicAdd(&den[d * HEADS + hh], expf(v - mx[d * HEADS + hh]));
    }
}

// Pass 3: agg[dst] += h[src] * alpha, alpha recomputed (no per-edge scratch)
__global__ void gat_edge_agg(const int* __restrict__ src,
                             const int* __restrict__ dst,
                             const float* __restrict__ alsrc,
                             const float* __restrict__ aldst,
                             const float* __restrict__ mx,
                             const float* __restrict__ den,
                             const float* __restrict__ h,
                             float* __restrict__ agg, int E) {
    int idx = blockIdx.x * blockDim.x + threadIdx.x;  // e*HEADS + head
    if (idx >= E * HEADS) return;
    const int e  = idx >> 2;
    const int hh = idx & 3;
    const int s = src[e], d = dst[e];
    const float v = leaky(alsrc[s * HEADS + hh] + aldst[d * HEADS + hh]);
    const float alpha = expf(v - mx[d * HEADS + hh]) / (den[d * HEADS + hh] + 1e-16f);
    const float* hp = h + s * HID + hh * DH;
    float* op       = agg + d * HID + hh * DH;
#pragma unroll
    for (int i = 0; i < DH; ++i)
        atomicAdd(op + i, hp[i] * alpha);
}

// out[i] = relu(out[i] + b[i % HID])  (in place)
__global__ void gat_finalize(float* __restrict__ x,
                             const float* __restrict__ b, int n) {
    int i = blockIdx.x * blockDim.x + threadIdx.x;
    if (i >= n) return;
    x[i] = fmaxf(x[i] + b[i % HID], 0.f);
}

// ---------------------------------------------------------------------------
// Driver
// ---------------------------------------------------------------------------
extern "C" void kernel_launch(void* const* d_in, const int* in_sizes, int n_in,
                              void* d_out, int out_size, void* d_ws, size_t ws_size,
                              hipStream_t stream) {
    const float* x       = (const float*)d_in[0];
    const int*   ei      = (const int*)d_in[1];
    const float* W1      = (const float*)d_in[2];
    const float* a_src1  = (const float*)d_in[3];
    const float* a_dst1  = (const float*)d_in[4];
    const float* b1      = (const float*)d_in[5];
    const float* W2      = (const float*)d_in[6];
    const float* a_src2  = (const float*)d_in[7];
    const float* a_dst2  = (const float*)d_in[8];
    const float* b2      = (const float*)d_in[9];

    const int N = in_sizes[0] / 128;   // 50000
    const int E = in_sizes[1] / 2;     // 800000
    const int* src = ei;
    const int* dst = ei + E;

    // workspace layout (floats)
    float* h     = (float*)d_ws;              // N*HID
    float* agg   = h     + (size_t)N * HID;   // N*HID
    float* alsrc = agg   + (size_t)N * HID;   // N*HEADS
    float* aldst = alsrc + (size_t)N * HEADS; // N*HEADS
    float* mx    = aldst + (size_t)N * HEADS; // N*HEADS
    float* den   = mx    + (size_t)N * HEADS; // N*HEADS
    float* out   = (float*)d_out;

    const int TB = 256;
    const int gemmWaves  = N / 16;                       // 3125 row tiles
    const int wavesPerBl = 8;                            // 256 threads
    const int gemmBlocks = (gemmWaves + wavesPerBl - 1) / wavesPerBl;
    dim3 gemmBlk(32 * wavesPerBl);
    const int nh   = N * HEADS;
    const int nhid = N * HID;
    const int eh   = E * HEADS;

    // ---------------- Layer 1 ----------------
    gat_gemm_wmma<<<gemmBlocks, gemmBlk, 0, stream>>>(x, W1, h, N, 128);
    gat_logits<<<(nh + TB - 1) / TB, TB, 0, stream>>>(h, a_src1, a_dst1, alsrc, aldst, N);
    fill_f32<<<(nh + TB - 1) / TB, TB, 0, stream>>>(mx, -INFINITY, nh);
    fill_f32<<<(nh + TB - 1) / TB, TB, 0, stream>>>(den, 0.f, nh);
    fill_f32<<<(nhid + TB - 1) / TB, TB, 0, stream>>>(agg, 0.f, nhid);
    gat_edge_max<<<(E + TB - 1) / TB, TB, 0, stream>>>(src, dst, alsrc, aldst, mx, E);
    gat_edge_sum<<<(E + TB - 1) / TB, TB, 0, stream>>>(src, dst, alsrc, aldst, mx, den, E);
    gat_edge_agg<<<(eh + TB - 1) / TB, TB, 0, stream>>>(src, dst, alsrc, aldst, mx, den, h, agg, E);
    gat_finalize<<<(nhid + TB - 1) / TB, TB, 0, stream>>>(agg, b1, nhid);   // agg now = x2

    // ---------------- Layer 2 ----------------
    gat_gemm_wmma<<<gemmBlocks, gemmBlk, 0, stream>>>(agg, W2, h, N, HID);
    gat_logits<<<(nh + TB - 1) / TB, TB, 0, stream>>>(h, a_src2, a_dst2, alsrc, aldst, N);
    fill_f32<<<(nh + TB - 1) / TB, TB, 0, stream>>>(mx, -INFINITY, nh);
    fill_f32<<<(nh + TB - 1) / TB, TB, 0, stream>>>(den, 0.f, nh);
    fill_f32<<<(nhid + TB - 1) / TB, TB, 0, stream>>>(out, 0.f, nhid);
    gat_edge_max<<<(E + TB - 1) / TB, TB, 0, stream>>>(src, dst, alsrc, aldst, mx, E);
    gat_edge_sum<<<(E + TB - 1) / TB, TB, 0, stream>>>(src, dst, alsrc, aldst, mx, den, E);
    gat_edge_agg<<<(eh + TB - 1) / TB, TB, 0, stream>>>(src, dst, alsrc, aldst, mx, den, h, out, E);
    gat_finalize<<<(nhid + TB - 1) / TB, TB, 0, stream>>>(out, b2, nhid);
}